// FullAttention_75127567942140
// MI455X (gfx1250) — compile-verified
//
#include <hip/hip_runtime.h>
#include <hip/hip_bf16.h>

// Problem constants (match reference)
#define BB 8
#define LL 1024
#define SS 1024
#define HH 16
#define EE 64
#define DD 64

#define LT 32          // L-tile rows per workgroup
#define SPAD 4         // LDS pad (floats); row stride 1028*4 B stays 16B-aligned

typedef __attribute__((ext_vector_type(2))) float v2f;
typedef __attribute__((ext_vector_type(4))) float v4f;
typedef __attribute__((ext_vector_type(8))) float v8f;

__global__ __launch_bounds__(256)
void FullAttention_75127567942140_kernel(const float* __restrict__ q,
                                         const float* __restrict__ k,
                                         const float* __restrict__ v,
                                         const float* __restrict__ bias,
                                         float* __restrict__ out,
                                         float* __restrict__ attn)
{
    // Scores tile: 32 rows x 1024 cols, padded stride (16B-aligned rows)
    __shared__ __align__(16) float sP[LT][SS + SPAD];
    __shared__ float sSum[LT];

    const int ltiles = LL / LT;                 // 32
    const int bid = blockIdx.x;
    const int lt  = bid % ltiles;
    const int h   = (bid / ltiles) % HH;
    const int b   = bid / (ltiles * HH);
    const int l0  = lt * LT;

    const int tid   = threadIdx.x;
    const int wave  = tid >> 5;                 // 0..7
    const int lane  = tid & 31;
    const int lhalf = lane >> 4;                // 0 or 1
    const int l16   = lane & 15;

    const int mt = wave >> 2;                   // 0..1  (16-row tile)
    const int nt = wave & 3;                    // 0..3  (16-col tile in 64-wide chunk)
    const int m0 = mt * 16;

    // Base pointers (row strides: q/k = H*E = 1024, v = H*D = 1024)
    const float* qbase = q + (size_t)b * LL * HH * EE + (size_t)h * EE;
    const float* kbase = k + (size_t)b * SS * HH * EE + (size_t)h * EE;
    const float* vbase = v + (size_t)b * SS * HH * DD + (size_t)h * DD;
    const float* biasbase = bias + (((size_t)b * HH + h) * LL + l0) * SS;

    const float scale = 0.125f;                 // 1/sqrt(64)

    // ---- Stage A-fragments of Q tile for this wave (rows m0..m0+15, all E) ----
    // A 16x4 f32 layout: lanes 0-15: v0=K0, v1=K1 ; lanes 16-31: v0=K2, v1=K3
    // Q is single-use per workgroup -> non-temporal b64 loads.
    v2f aQ[16];
    {
        const float* qr = qbase + (size_t)(l0 + m0 + l16) * (HH * EE) + 2 * lhalf;
        #pragma unroll
        for (int i = 0; i < 16; ++i)
            aQ[i] = __builtin_nontemporal_load((const v2f*)&qr[4 * i]);
    }

    // ---- Phase 1: scores = scale * (Q K^T + bias) into LDS ----
    for (int sc = 0; sc < SS; sc += 64) {
        const int n0 = sc + nt * 16;
        v8f c = {};
        // B 4x16 layout (K^T): lanes 0-15: v0=K0,v1=K1 ; lanes 16-31: v0=K2,v1=K3
        // K stays regular-temporal: heavy cross-workgroup reuse in L2.
        const float* kr = kbase + (size_t)(n0 + l16) * (HH * EE) + 2 * lhalf;
        #pragma unroll
        for (int i = 0; i < 16; ++i) {
            const v2f bK = *(const v2f*)&kr[4 * i];
            c = __builtin_amdgcn_wmma_f32_16x16x4_f32(
                    false, aQ[i], false, bK, (short)0, c, false, false);
        }
        // C/D layout: VGPR j -> M = m0 + j + 8*lhalf, N = n0 + l16
        // bias is a 537MB single-use stream -> non-temporal loads.
        #pragma unroll
        for (int j = 0; j < 8; ++j) {
            const int row = m0 + j + 8 * lhalf;
            const int col = n0 + l16;
            const float bv = __builtin_nontemporal_load(
                                 &biasbase[(size_t)row * SS + col]);
            sP[row][col] = (c[j] + bv) * scale;
        }
    }
    __syncthreads();

    // ---- Phase 2: row softmax, vectorized b128 LDS traffic ----
    for (int r = wave; r < LT; r += 8) {
        v4f* rowp = (v4f*)&sP[r][0];            // 256 float4 per row
        float mx = -__builtin_inff();
        for (int c4 = lane; c4 < SS / 4; c4 += 32) {
            const v4f t = rowp[c4];
            mx = fmaxf(mx, fmaxf(fmaxf(t.x, t.y), fmaxf(t.z, t.w)));
        }
        #pragma unroll
        for (int off = 16; off > 0; off >>= 1)
            mx = fmaxf(mx, __shfl_xor(mx, off, 32));
        float sum = 0.0f;
        for (int c4 = lane; c4 < SS / 4; c4 += 32) {
            v4f t = rowp[c4];
            t.x = __expf(t.x - mx);
            t.y = __expf(t.y - mx);
            t.z = __expf(t.z - mx);
            t.w = __expf(t.w - mx);
            rowp[c4] = t;
            sum += (t.x + t.y) + (t.z + t.w);
        }
        #pragma unroll
        for (int off = 16; off > 0; off >>= 1)
            sum += __shfl_xor(sum, off, 32);
        if (lane == 0) sSum[r] = sum;
    }
    __syncthreads();

    // ---- Phase 3a: normalized attn write, coalesced b128 NT stores ----
    {
        float* attnbase = attn + (((size_t)b * HH + h) * LL + l0) * SS;
        for (int f4 = tid; f4 < LT * SS / 4; f4 += 256) {
            const int r  = f4 >> 8;             // 256 float4 per row
            const int c4 = f4 & 255;
            const v4f p  = ((const v4f*)&sP[r][0])[c4];
            const float inv = 1.0f / sSum[r];
            v4f a; a.x = p.x * inv; a.y = p.y * inv; a.z = p.z * inv; a.w = p.w * inv;
            __builtin_nontemporal_store(a, (v4f*)&attnbase[(size_t)r * SS + 4 * c4]);
        }
    }

    // ---- Phase 3b: out = (P / rowsum) @ V via WMMA, wave tile 16x16 of D ----
    {
        const int n0d = nt * 16;
        v8f o = {};
        for (int kk = 0; kk < SS; kk += 4) {
            // A from LDS P (b64): lane<16 -> K = kk+0..1 ; lane>=16 -> K = kk+2..3
            const v2f aP = *(const v2f*)&sP[m0 + l16][kk + 2 * lhalf];
            // B from V (regular-temporal: L2-resident, reused by all L-tiles):
            // lanes 0-15: v0=V[kk+0][n], v1=V[kk+1][n]; lanes 16-31: kk+2/kk+3
            const float* vr = vbase + (size_t)(kk + 2 * lhalf) * (HH * DD) + n0d + l16;
            v2f bV;
            bV.x = vr[0];
            bV.y = vr[HH * DD];
            o = __builtin_amdgcn_wmma_f32_16x16x4_f32(
                    false, aP, false, bV, (short)0, o, false, false);
        }
        float* outbase = out + (((size_t)b * LL + l0) * HH + h) * DD;
        #pragma unroll
        for (int j = 0; j < 8; ++j) {
            const int row = m0 + j + 8 * lhalf;
            const int col = n0d + l16;
            const float ov = o[j] * (1.0f / sSum[row]);
            __builtin_nontemporal_store(ov, &outbase[(size_t)row * (HH * DD) + col]);
        }
    }
}

extern "C" void kernel_launch(void* const* d_in, const int* in_sizes, int n_in,
                              void* d_out, int out_size, void* d_ws, size_t ws_size,
                              hipStream_t stream) {
    (void)in_sizes; (void)n_in; (void)d_ws; (void)ws_size; (void)out_size;
    const float* q    = (const float*)d_in[0];
    const float* k    = (const float*)d_in[1];
    const float* v    = (const float*)d_in[2];
    const float* bias = (const float*)d_in[3];
    float* out  = (float*)d_out;                                   // [B,L,H,D]
    float* attn = (float*)d_out + (size_t)BB * LL * HH * DD;       // [B,H,L,S]

    const int blocks = BB * HH * (LL / LT);                        // 4096
    FullAttention_75127567942140_kernel<<<blocks, 256, 0, stream>>>(
        q, k, v, bias, out, attn);
}